// LearnableTSM_38646115730016
// MI455X (gfx1250) — compile-verified
//
#include <hip/hip_runtime.h>

// Temporal Shift Module (fixed shift, SHIFT_RATIO=0.5, SHIFT_GROUPS=2)
// x[B=4, T=16, H=64, W=64, C=256] float32.
//   c in [  0, 64)  : out[b,t] = x[b,t+1]  (zero at t == T-1)
//   c in [ 64,128)  : out[b,t] = x[b,t-1]  (zero at t == 0)
//   c in [128,256)  : out[b,t] = x[b,t]
//
// Pure streaming op: 256 MiB in + 256 MiB out @ 23.3 TB/s ~= 23 us floor.
// Each thread moves a 64 B chunk (4 x float4) of one channel group so the
// shift direction and edge predicate are computed once per 64 B, keeping
// the device-wide instruction issue rate well above what 23.3 TB/s needs.
// Non-temporal hints both ways: working set (512 MiB) >> 192 MB L2 and
// every element is touched exactly once.

typedef float f4 __attribute__((ext_vector_type(4)));

#define TSM_T         16
#define TSM_STRIDE_T4 262144    // H*W*C/4 : float4 stride between t-slices
#define TSM_PF_DIST4  65536     // prefetch distance in float4s (~1 MiB)

__global__ __launch_bounds__(256) void tsm_shift_kernel(
    const f4* __restrict__ in, f4* __restrict__ out, int nchunks) {
  int cidx = blockIdx.x * 256 + threadIdx.x;   // chunk = 4 consecutive float4
  if (cidx >= nchunks) return;

  int cb = cidx & 15;                 // 16 chunks per (b,t,h,w) position
  int t  = (cidx >> 16) & (TSM_T - 1);

  // Chunk groups: [0,4)->fwd(+1), [4,8)->bwd(-1), [8,16)->copy(0).
  // Group boundaries are chunk-aligned, so dt is uniform within a chunk.
  int dt = (cb < 4) ? 1 : ((cb < 8) ? -1 : 0);
  int st = t + dt;

  int base = cidx << 2;               // float4 index of chunk start

  // Prefetch the copy stream ~1 MiB ahead into GL2 (guarded).
  if (base + TSM_PF_DIST4 < (nchunks << 2))
    __builtin_prefetch((const void*)(in + base + TSM_PF_DIST4), 0, 0);

  f4 v0 = (f4)(0.0f), v1 = v0, v2 = v0, v3 = v0;
  if ((unsigned)st < (unsigned)TSM_T) {
    const f4* src = in + (base + dt * TSM_STRIDE_T4);
    v0 = __builtin_nontemporal_load(src + 0);
    v1 = __builtin_nontemporal_load(src + 1);
    v2 = __builtin_nontemporal_load(src + 2);
    v3 = __builtin_nontemporal_load(src + 3);
  }
  f4* dst = out + base;
  __builtin_nontemporal_store(v0, dst + 0);
  __builtin_nontemporal_store(v1, dst + 1);
  __builtin_nontemporal_store(v2, dst + 2);
  __builtin_nontemporal_store(v3, dst + 3);
}

extern "C" void kernel_launch(void* const* d_in, const int* in_sizes, int n_in,
                              void* d_out, int out_size, void* d_ws, size_t ws_size,
                              hipStream_t stream) {
  const f4* in = (const f4*)d_in[0];
  f4* out = (f4*)d_out;
  int nchunks = in_sizes[0] / 16;               // 4,194,304 64-byte chunks
  int blocks = (nchunks + 255) / 256;           // 16,384 blocks of 256 (8 waves)
  tsm_shift_kernel<<<blocks, 256, 0, stream>>>(in, out, nchunks);
}